// SequentialTransductionUnitJagged_30356828848482
// MI455X (gfx1250) — compile-verified
//
#include <hip/hip_runtime.h>
#include <hip/hip_bf16.h>
#include <math.h>
#include <stdint.h>

// MI455X (gfx1250) fp32 WMMA implementation of the jagged HSTU block.
// GEMMs use V_WMMA_F32_16X16X4_F32; attention K/V staging uses the Tensor
// Data Mover (TENSOR_LOAD_TO_LDS) double-buffered against the WMMA loop.

typedef __attribute__((ext_vector_type(2))) float v2f;
typedef __attribute__((ext_vector_type(8))) float v8f;
typedef __attribute__((ext_vector_type(4))) unsigned int u32x4;
typedef __attribute__((ext_vector_type(8))) int i32x8;
typedef __attribute__((ext_vector_type(4))) int i32x4;

#define TT   6400     // total jagged tokens
#define DD   512      // model dim
#define HID  1024     // fused uvqk output dim (u|v|q|k, 256 each)
#define BB   4
#define NN   2048     // dense sequence length (for timestamps & 1/N scale)
#define NH   4        // heads
#define DH   64       // head dim (== LH == AH)
#define NBUCK 128
#define KVS  260      // LDS row stride: 256 data + 4 pad DWORDs (TDM pad does
                      // this in HW) -> frag reads hit banks (4*cN+2g+k)%64

#if defined(__has_builtin)
#if __has_builtin(__builtin_amdgcn_tensor_load_to_lds)
#define USE_TDM 1
#endif
#endif

__device__ __forceinline__ float silu_f(float x) {
    return x / (1.0f + expf(-x));
}

__device__ __forceinline__ v8f wmma4(v2f a, v2f b, v8f c) {
    return __builtin_amdgcn_wmma_f32_16x16x4_f32(false, a, false, b, (short)0, c,
                                                 false, false);
}

#ifdef USE_TDM
// Issue one TDM 2D tile load: 16 rows x 256 fp32, global row stride 1024 fp32,
// LDS destination padded by 4 DWORDs every 256 DWORDs (=> 260-float rows).
// D# packed per CDNA5 ISA 8.3/8.4; this toolchain's builtin takes 6 args:
// (group0 v4u, group1 v8i, group2 v4i, group3 v4i, v8i (zero), cpol).
__device__ __forceinline__ void tdm_load_tile(unsigned lds_off, const float* gptr) {
    const unsigned long long ga = (unsigned long long)(uintptr_t)gptr;
    u32x4 g0;
    g0.x = 1u;                                        // count=1, user mode
    g0.y = lds_off;                                   // lds_addr (bytes)
    g0.z = (unsigned)(ga & 0xffffffffu);              // global_addr[31:0]
    g0.w = (unsigned)((ga >> 32) & 0x1ffffffu)        // global_addr[56:32]
         | (2u << 30);                                // type=2 ("image")
    i32x8 g1;
    g1[0] = (int)((2u << 16)      // data_size = 4B
                | (1u << 20)      // pad_enable
                | (7u << 22)      // pad_interval: every 256 DWORDs
                | (3u << 25));    // pad_amount: 4 DWORDs
    g1[1] = (int)(256u << 16);    // atomic_addr=0 | tensor_dim0[15:0] = 256
    g1[2] = (int)(16u << 16);     // tensor_dim0[31:16]=0 | tensor_dim1[15:0]=16
    g1[3] = (int)(256u << 16);    // tensor_dim1[31:16]=0 | tile_dim0 = 256
    g1[4] = 16;                   // tile_dim1 = 16 | tile_dim2 = 0
    g1[5] = 1024;                 // tensor_dim0_stride[31:0] = 1024 elements
    g1[6] = 0;                    // stride0[47:32]=0 | dim1_stride lo = 0
    g1[7] = 0;
    const i32x4 z4 = {0, 0, 0, 0};                    // groups 2/3: unused (2D)
    const i32x8 z8 = {0, 0, 0, 0, 0, 0, 0, 0};        // unprobed trailing arg
    __builtin_amdgcn_tensor_load_to_lds(g0, g1, z4, z4, z8, 0);
}
#endif

// ---------------------------------------------------------------------------
// Kernel 1: h = silu( layernorm(x) @ uvqk )   [T,512] x [512,1024] -> [T,1024]
// Block: 256 threads (8 waves). grid = (T/16, 1024/128).
// uvqk (2 MB) stays resident in the 192 MB L2 across all 400 row tiles.
// ---------------------------------------------------------------------------
__global__ __launch_bounds__(256)
void k_ln_proj(const float* __restrict__ x, const float* __restrict__ uvqk,
               float* __restrict__ h) {
    __shared__ float lnx[16 * 516];          // stride 516: conflict-free A reads
    const int tile = blockIdx.x;
    const int tid  = threadIdx.x;

    {   // LayerNorm: 16 threads per row, 32 elements each
        const int row = tid >> 4;
        const int l16 = tid & 15;
        const float* xr = x + (size_t)(tile * 16 + row) * DD;
        float s = 0.f, ss = 0.f;
        float vals[32];
#pragma unroll
        for (int i = 0; i < 32; ++i) {
            float v = xr[l16 + 16 * i];
            vals[i] = v; s += v; ss += v * v;
        }
#pragma unroll
        for (int m = 1; m <= 8; m <<= 1) {   // reduce within 16-lane halves
            s  += __shfl_xor(s,  m, 32);
            ss += __shfl_xor(ss, m, 32);
        }
        const float mean = s * (1.0f / DD);
        const float var  = ss * (1.0f / DD) - mean * mean;  // biased var
        const float rstd = rsqrtf(var + 1e-6f);
#pragma unroll
        for (int i = 0; i < 32; ++i)
            lnx[row * 516 + l16 + 16 * i] = (vals[i] - mean) * rstd;
    }
    __syncthreads();

    const int wave = tid >> 5;
    const int lane = tid & 31;
    const int cN   = lane & 15;
    const int g    = lane >> 4;
    const int col0 = blockIdx.y * 128 + wave * 16;

    v8f acc = {};
    const float* bp = uvqk + col0 + cN;
#pragma unroll 4
    for (int k0 = 0; k0 < DD; k0 += 4) {
        v2f a = *(const v2f*)&lnx[cN * 516 + k0 + 2 * g];      // A[M][k0+2g..]
        v2f b;
        b.x = bp[(size_t)(k0 + 2 * g) * HID];                  // B[k][col]
        b.y = bp[(size_t)(k0 + 2 * g + 1) * HID];
        acc = wmma4(a, b, acc);
    }
    float* hp = h + (size_t)(tile * 16 + 8 * g) * HID + col0 + cN;
#pragma unroll
    for (int j = 0; j < 8; ++j)
        hp[(size_t)j * HID] = silu_f(acc[j]);
}

// ---------------------------------------------------------------------------
// Kernel 2: jagged causal attention with time-bucket bias + SiLU scoring.
// Block = 128 threads (4 waves, one head each). grid = (T/16).
// K/V slabs (16 x 256 fp32) are DMA'd into LDS by the Tensor Data Mover,
// double-buffered: tile mt+1 streams in while WMMAs chew on tile mt.
// h layout per token row (stride 1024): [u | v | q | k], 256 floats each.
// ---------------------------------------------------------------------------
__global__ __launch_bounds__(128)
void k_attn(const float* __restrict__ h, const int* __restrict__ ts,
            const float* __restrict__ ts_w, const int* __restrict__ offsets,
            float* __restrict__ attn) {
    __shared__ float sKV[2][2][16 * KVS];    // [buffer][K=0 / V=1]
    __shared__ float sS[4 * 16 * 18];        // per-wave score tile (C->A relayout)
    const int tid  = threadIdx.x;
    const int lane = tid & 31;
    const int wave = tid >> 5;               // == head
    const int head = wave;
    const int cN   = lane & 15;
    const int g    = lane >> 4;

    // map global row tile -> (batch, local n0); all offsets are multiples of 16
    const int row0 = blockIdx.x * 16;
    int b = 0;
#pragma unroll
    for (int i = 1; i < BB; ++i)
        if (row0 >= offsets[i]) b = i;
    const int offB = offsets[b];
    const int n0   = row0 - offB;
    const int mtiles = n0 / 16 + 1;          // causal: m-tile start <= n0

#ifndef USE_TDM
    const int srow = tid >> 6;               // cooperative fallback staging map
    const int sc4  = (tid & 63) * 4;
#endif

    // ---- stage Q slab into buffer 0 (K slot), build A fragments ----
#ifdef USE_TDM
    if (wave == 0) {
        tdm_load_tile((unsigned)(unsigned long long)(uintptr_t)&sKV[0][0][0],
                      h + (size_t)(offB + n0) * HID + 512);
        __builtin_amdgcn_s_wait_tensorcnt(0);
    }
#else
#pragma unroll
    for (int i = 0; i < 8; ++i) {
        const int r = srow + 2 * i;
        *(float4*)&sKV[0][0][r * KVS + sc4] =
            *(const float4*)&h[(size_t)(offB + n0 + r) * HID + 512 + sc4];
    }
#endif
    __syncthreads();
    v2f aq[16];
#pragma unroll
    for (int i = 0; i < 16; ++i)
        aq[i] = *(const v2f*)&sKV[0][0][cN * KVS + head * DH + 4 * i + 2 * g];
    __syncthreads();                          // buffer 0 about to be reused

    // row-side timestamps are m-invariant: hoist out of the m-loop
    const int* tsb = ts + b * NN;
    int tn[8];
#pragma unroll
    for (int j = 0; j < 8; ++j)
        tn[j] = tsb[n0 + j + 8 * g];

#ifdef USE_TDM
    if (wave == 0) {                          // prefetch tile 0 into buffer 0
        tdm_load_tile((unsigned)(unsigned long long)(uintptr_t)&sKV[0][0][0],
                      h + (size_t)offB * HID + 768);
        tdm_load_tile((unsigned)(unsigned long long)(uintptr_t)&sKV[0][1][0],
                      h + (size_t)offB * HID + 256);
    }
#endif

    float* mysS = &sS[wave * 16 * 18];
    v8f o0 = {}, o1 = {}, o2 = {}, o3 = {};

    for (int mt = 0; mt < mtiles; ++mt) {
        const int m0  = mt * 16;
        const int cur = mt & 1;

#ifdef USE_TDM
        // issue DMA for tile mt+1, then wait for tile mt (TDM ops are in-order)
        if (wave == 0) {
            if (mt + 1 < mtiles) {
                const size_t rb = (size_t)(offB + m0 + 16) * HID;
                tdm_load_tile(
                    (unsigned)(unsigned long long)(uintptr_t)&sKV[cur ^ 1][0][0],
                    h + rb + 768);
                tdm_load_tile(
                    (unsigned)(unsigned long long)(uintptr_t)&sKV[cur ^ 1][1][0],
                    h + rb + 256);
                __builtin_amdgcn_s_wait_tensorcnt(2);
            } else {
                __builtin_amdgcn_s_wait_tensorcnt(0);
            }
        }
#else
#pragma unroll
        for (int i = 0; i < 8; ++i) {
            const int r = srow + 2 * i;
            const size_t rb = (size_t)(offB + m0 + r) * HID;
            *(float4*)&sKV[cur][0][r * KVS + sc4] = *(const float4*)&h[rb + 768 + sc4];
            *(float4*)&sKV[cur][1][r * KVS + sc4] = *(const float4*)&h[rb + 256 + sc4];
        }
#endif
        __syncthreads();                      // tile mt visible to all waves

        // ---- S = Q @ K^T (16x16), K-reduction of 64 in 16 WMMA steps ----
        v8f S = {};
        const float* kb = &sKV[cur][0][cN * KVS + head * DH]; // B[k][m]=K[m0+cN][k]
#pragma unroll
        for (int i = 0; i < 16; ++i) {
            v2f bk = *(const v2f*)&kb[4 * i + 2 * g];
            S = wmma4(aq[i], bk, S);
        }

        // ---- bias + silu/N + causal mask, C/D layout -> LDS (A layout src) ----
        const int m   = m0 + cN;
        const int tsm = tsb[m];
#pragma unroll
        for (int j = 0; j < 8; ++j) {
            float s = S[j];
            const float delta = fabsf((float)(tn[j] - tsm));
            int bucket = (int)floorf(log1pf(delta));
            bucket = bucket < 0 ? 0 : (bucket > NBUCK ? NBUCK : bucket);
            s += ts_w[bucket];
            s = silu_f(s) * (1.0f / NN);
            if (m > n0 + j + 8 * g) s = 0.0f; // tril mask (lane select, no EXEC div)
            mysS[(j + 8 * g) * 18 + cN] = s;
        }
        // mysS is wave-private; same-wave LDS ops are in-order -> no barrier

        // ---- O[16,64] += S @ V : 4 col-chunks x 4 K-steps of WMMA ----
        const float* vb = &sKV[cur][1][head * DH];
#pragma unroll
        for (int k0 = 0; k0 < 16; k0 += 4) {
            v2f a = *(const v2f*)&mysS[cN * 18 + k0 + 2 * g];  // A[M=cN][k]
            const float* vr0 = vb + (k0 + 2 * g) * KVS;
            const float* vr1 = vr0 + KVS;
            v2f bv;
            bv.x = vr0[cN];      bv.y = vr1[cN];      o0 = wmma4(a, bv, o0);
            bv.x = vr0[16 + cN]; bv.y = vr1[16 + cN]; o1 = wmma4(a, bv, o1);
            bv.x = vr0[32 + cN]; bv.y = vr1[32 + cN]; o2 = wmma4(a, bv, o2);
            bv.x = vr0[48 + cN]; bv.y = vr1[48 + cN]; o3 = wmma4(a, bv, o3);
        }
        __syncthreads();          // all waves done with tile mt before its buffer
                                  // is re-targeted by the DMA of tile mt+2
    }

    // ---- store attn tile: [T, 256], col = head*64 + chunk*16 + cN ----
    float* op = attn + (size_t)(offB + n0 + 8 * g) * 256 + head * DH + cN;
#pragma unroll
    for (int j = 0; j < 8; ++j) {
        op[(size_t)j * 256 +  0] = o0[j];
        op[(size_t)j * 256 + 16] = o1[j];
        op[(size_t)j * 256 + 32] = o2[j];
        op[(size_t)j * 256 + 48] = o3[j];
    }
}

// ---------------------------------------------------------------------------
// Kernel 3: out = (u * layernorm(attn)) @ o_w^T + o_b + x
// Block: 256 threads (8 waves). grid = (T/16, 512/128).
// ---------------------------------------------------------------------------
__global__ __launch_bounds__(256)
void k_out(const float* __restrict__ h, const float* __restrict__ attn,
           const float* __restrict__ o_w, const float* __restrict__ o_b,
           const float* __restrict__ x, float* __restrict__ out) {
    __shared__ float oin[16 * 260];          // stride 260: aligned, conflict-free
    const int tile = blockIdx.x;
    const int tid  = threadIdx.x;

    {   // LN(attn) * u, staged in LDS: 16 threads per row, 16 elems each
        const int row = tid >> 4;
        const int l16 = tid & 15;
        const int t   = tile * 16 + row;
        const float* ar = attn + (size_t)t * 256;
        const float* ur = h + (size_t)t * HID;       // u = h[:, 0:256]
        float s = 0.f, ss = 0.f;
        float vals[16];
#pragma unroll
        for (int i = 0; i < 16; ++i) {
            float v = ar[l16 + 16 * i];
            vals[i] = v; s += v; ss += v * v;
        }
#pragma unroll
        for (int m = 1; m <= 8; m <<= 1) {
            s  += __shfl_xor(s,  m, 32);
            ss += __shfl_xor(ss, m, 32);
        }
        const float mean = s * (1.0f / 256.0f);
        const float var  = ss * (1.0f / 256.0f) - mean * mean;
        const float rstd = rsqrtf(var + 1e-6f);
#pragma unroll
        for (int i = 0; i < 16; ++i)
            oin[row * 260 + l16 + 16 * i] =
                ur[l16 + 16 * i] * (vals[i] - mean) * rstd;
    }
    __syncthreads();

    const int wave = tid >> 5;
    const int lane = tid & 31;
    const int cN   = lane & 15;
    const int g    = lane >> 4;
    const int col0 = blockIdx.y * 128 + wave * 16;

    const float* bp = o_w + (size_t)(col0 + cN) * 256;  // B[k][n] = o_w[n][k]
    v8f acc = {};
#pragma unroll 4
    for (int k0 = 0; k0 < 256; k0 += 4) {
        v2f a = *(const v2f*)&oin[cN * 260 + k0 + 2 * g];
        v2f b = *(const v2f*)&bp[k0 + 2 * g];            // consecutive -> b64
        acc = wmma4(a, b, acc);
    }
    const float bias = o_b[col0 + cN];
    const int t0 = tile * 16 + 8 * g;
#pragma unroll
    for (int j = 0; j < 8; ++j) {
        const int t = t0 + j;
        out[(size_t)t * DD + col0 + cN] =
            acc[j] + bias + x[(size_t)t * DD + col0 + cN];
    }
}

// ---------------------------------------------------------------------------
extern "C" void kernel_launch(void* const* d_in, const int* in_sizes, int n_in,
                              void* d_out, int out_size, void* d_ws, size_t ws_size,
                              hipStream_t stream) {
    (void)in_sizes; (void)n_in; (void)out_size; (void)ws_size;
    const float* x      = (const float*)d_in[0];
    const float* uvqk   = (const float*)d_in[1];
    const float* o_w    = (const float*)d_in[2];
    const float* o_b    = (const float*)d_in[3];
    const float* ts_w   = (const float*)d_in[4];
    const int*   all_ts = (const int*)d_in[5];
    // d_in[6] invalid_attn_mask: tril(ones) -> causal mask applied analytically
    const int*   offs   = (const int*)d_in[7];
    // d_in[8], d_in[9]: token_batch/token_pos unused (offsets suffice; tiles
    // never straddle batches since all offsets are multiples of 16)

    float* h    = (float*)d_ws;                      // [T, 1024]  ~26.2 MB
    float* attn = h + (size_t)TT * HID;              // [T, 256]   ~6.6 MB
    float* out  = (float*)d_out;                     // [T, 512]

    dim3 g1(TT / 16, HID / 128), b1(256);
    hipLaunchKernelGGL(k_ln_proj, g1, b1, 0, stream, x, uvqk, h);

    dim3 g2(TT / 16), b2(128);
    hipLaunchKernelGGL(k_attn, g2, b2, 0, stream, h, all_ts, ts_w, offs, attn);

    dim3 g3(TT / 16, DD / 128), b3(256);
    hipLaunchKernelGGL(k_out, g3, b3, 0, stream, h, attn, o_w, o_b, x, out);
}